// AttnDecoderRNN_82274393522645
// MI455X (gfx1250) — compile-verified
//
#include <hip/hip_runtime.h>
#include <hip/hip_bf16.h>
#include <math.h>

// ---------------------------------------------------------------------------
// AttnDecoderRNN single step, MI455X (gfx1250, wave32).
// All heavy ops are batch-1 GEMVs -> HBM bandwidth bound (~510 MB, ~22us floor).
// Row-major GEMVs use V_WMMA_F32_16X16X4_F32 (fp32 matrix pipe, one wave per
// 16-row tile, x broadcast into all B columns). Column GEMVs use split-K
// coalesced VALU passes + reduce. Softmax / log-softmax are block reductions.
// ---------------------------------------------------------------------------

#define Hn 1024
#define Vn 50257
#define Sn 4096

typedef float v2f __attribute__((ext_vector_type(2)));
typedef float v8f __attribute__((ext_vector_type(8)));

// ---------------- WMMA row-GEMV:  y[r] = sum_k W[r,k]*x[k] (+ bias[r]) -------
// W: rows x K row-major, K % 4 == 0. One wave (32 lanes) per 16-row tile.
__global__ void wmma_gemv_f32(const float* __restrict__ W,
                              const float* __restrict__ x,
                              const float* __restrict__ bias,
                              float* __restrict__ y,
                              int rows, int K) {
  int gtid = blockIdx.x * blockDim.x + threadIdx.x;
  int wave = gtid >> 5;
  int lane = threadIdx.x & 31;
  int m0 = wave * 16;
  if (m0 >= rows) return;          // whole-wave uniform exit: EXEC stays full

  int hw   = lane >> 4;            // 0 = lanes 0-15, 1 = lanes 16-31
  int mrow = m0 + (lane & 15);
  int mclamp = (mrow < rows) ? mrow : (rows - 1);
  const float* wrow = W + (size_t)mclamp * (size_t)K + (size_t)(hw * 2);

  v8f acc = {0.f, 0.f, 0.f, 0.f, 0.f, 0.f, 0.f, 0.f};

  for (int k = 0; k < K; k += 4) {
    // A 16x4 tile, ISA layout: lane<16 holds (K=k,k+1) of row M=lane,
    //                          lane>=16 holds (K=k+2,k+3) of row M=lane-16.
    v2f a = *(const v2f*)(wrow + k);
    // B 4x16 with every column = x  =>  all D columns equal y.
    v2f b;
    b.x = x[k + hw * 2 + 0];
    b.y = x[k + hw * 2 + 1];
    acc = __builtin_amdgcn_wmma_f32_16x16x4_f32(
        /*neg_a=*/false, a, /*neg_b=*/false, b,
        /*c_mod=*/(short)0, acc, /*reuse_a=*/false, /*reuse_b=*/false);
  }

  // D layout: VGPR j: lanes 0-15 -> M=j, lanes 16-31 -> M=8+j (all N equal).
  if ((lane & 15) == 0) {
    int base = m0 + hw * 8;
#pragma unroll
    for (int j = 0; j < 8; ++j) {
      int r = base + j;
      if (r < rows) {
        float v = acc[j];
        if (bias) v += bias[r];
        y[r] = v;
      }
    }
  }
}

// ---------------- column GEMV partials: part[b][i] = sum_{j in chunk} w[j]*A[j,i]
// A: M x N row-major, blockDim.x == N lanes are coalesced along i.
__global__ void col_gemv_partial(const float* __restrict__ A,
                                 const float* __restrict__ w,
                                 float* __restrict__ part,
                                 int M, int N) {
  int i = threadIdx.x;
  int chunk = (M + gridDim.x - 1) / gridDim.x;
  int j0 = blockIdx.x * chunk;
  int j1 = j0 + chunk; if (j1 > M) j1 = M;
  float s = 0.f;
  for (int j = j0; j < j1; ++j)
    s = fmaf(w[j], A[(size_t)j * (size_t)N + i], s);
  part[(size_t)blockIdx.x * (size_t)N + i] = s;
}

__global__ void reduce_parts(const float* __restrict__ part,
                             float* __restrict__ out,
                             float* __restrict__ out2,
                             int nparts, int N) {
  int i = blockIdx.x * blockDim.x + threadIdx.x;
  if (i >= N) return;
  float s = 0.f;
  for (int p = 0; p < nparts; ++p) s += part[(size_t)p * (size_t)N + i];
  out[i] = s;
  if (out2) out2[i] = s;
}

// ---------------- misc elementwise kernels ----------------------------------
__global__ void embed_concat(const int* __restrict__ word,
                             const float* __restrict__ emb,
                             const float* __restrict__ last_ctx,
                             float* __restrict__ rnn_in) {
  int i = blockIdx.x * blockDim.x + threadIdx.x;
  if (i < Hn)            rnn_in[i] = emb[(size_t)word[0] * Hn + i];
  else if (i < 2 * Hn)   rnn_in[i] = last_ctx[i - Hn];
}

__device__ __forceinline__ float sigmoidf(float x) {
  return 1.0f / (1.0f + expf(-x));
}

__global__ void gru_gate(const float* __restrict__ gi,
                         const float* __restrict__ gh,
                         const float* __restrict__ hprev,
                         float* __restrict__ h_ws,
                         float* __restrict__ h_out) {
  int i = blockIdx.x * blockDim.x + threadIdx.x;
  if (i >= Hn) return;
  float r = sigmoidf(gi[i] + gh[i]);
  float z = sigmoidf(gi[Hn + i] + gh[Hn + i]);
  float n = tanhf(gi[2 * Hn + i] + r * gh[2 * Hn + i]);
  float h = (1.0f - z) * n + z * hprev[i];
  h_ws[i]  = h;
  h_out[i] = h;
}

// softmax over n (<= few K), one block of 1024 threads
__global__ void softmax_kernel(const float* __restrict__ e,
                               float* __restrict__ attn_ws,
                               float* __restrict__ attn_out,
                               int n) {
  __shared__ float sb[1024];
  int t = threadIdx.x, nt = blockDim.x;
  float m = -3.4e38f;
  for (int i = t; i < n; i += nt) m = fmaxf(m, e[i]);
  sb[t] = m; __syncthreads();
  for (int o = nt >> 1; o; o >>= 1) { if (t < o) sb[t] = fmaxf(sb[t], sb[t + o]); __syncthreads(); }
  float M = sb[0]; __syncthreads();
  float s = 0.f;
  for (int i = t; i < n; i += nt) s += expf(e[i] - M);
  sb[t] = s; __syncthreads();
  for (int o = nt >> 1; o; o >>= 1) { if (t < o) sb[t] += sb[t + o]; __syncthreads(); }
  float inv = 1.0f / sb[0];
  for (int i = t; i < n; i += nt) {
    float a = expf(e[i] - M) * inv;
    attn_ws[i]  = a;
    attn_out[i] = a;
  }
}

// one block: scal[0]=max(x), scal[1]=log(sum exp(x-max))
__global__ void logsumexp_kernel(const float* __restrict__ x,
                                 float* __restrict__ scal, int n) {
  __shared__ float sb[1024];
  int t = threadIdx.x, nt = blockDim.x;
  float m = -3.4e38f;
  for (int i = t; i < n; i += nt) m = fmaxf(m, x[i]);
  sb[t] = m; __syncthreads();
  for (int o = nt >> 1; o; o >>= 1) { if (t < o) sb[t] = fmaxf(sb[t], sb[t + o]); __syncthreads(); }
  float M = sb[0]; __syncthreads();
  float s = 0.f;
  for (int i = t; i < n; i += nt) s += expf(x[i] - M);
  sb[t] = s; __syncthreads();
  for (int o = nt >> 1; o; o >>= 1) { if (t < o) sb[t] += sb[t + o]; __syncthreads(); }
  if (t == 0) { scal[0] = M; scal[1] = logf(sb[0]); }
}

__global__ void log_softmax_write(const float* __restrict__ logits,
                                  const float* __restrict__ scal,
                                  float* __restrict__ out, int n) {
  int i = blockIdx.x * blockDim.x + threadIdx.x;
  if (i < n) out[i] = logits[i] - scal[0] - scal[1];
}

// ---------------- host launcher ---------------------------------------------
static inline void launch_gemv(const float* W, const float* x, const float* b,
                               float* y, int rows, int K, hipStream_t stream) {
  int waves   = (rows + 15) / 16;
  int threads = waves * 32;
  int block   = 256;
  int grid    = (threads + block - 1) / block;
  wmma_gemv_f32<<<grid, block, 0, stream>>>(W, x, b, y, rows, K);
}

extern "C" void kernel_launch(void* const* d_in, const int* in_sizes, int n_in,
                              void* d_out, int out_size, void* d_ws, size_t ws_size,
                              hipStream_t stream) {
  const int*   word     = (const int*)  d_in[0];
  const float* last_ctx = (const float*)d_in[1];
  const float* last_hid = (const float*)d_in[2];   // [2][1][H]
  const float* enc      = (const float*)d_in[3];   // [S][1][H] == S x H row-major
  const float* emb      = (const float*)d_in[4];
  const float* W_ih0    = (const float*)d_in[5];
  const float* W_hh0    = (const float*)d_in[6];
  const float* b_ih0    = (const float*)d_in[7];
  const float* b_hh0    = (const float*)d_in[8];
  const float* W_ih1    = (const float*)d_in[9];
  const float* W_hh1    = (const float*)d_in[10];
  const float* b_ih1    = (const float*)d_in[11];
  const float* b_hh1    = (const float*)d_in[12];
  const float* W_attn   = (const float*)d_in[13];
  // d_in[14] = b_attn: adds a softmax-invariant constant to energies -> skipped
  const float* W_out    = (const float*)d_in[15];
  const float* b_out    = (const float*)d_in[16];

  float* out = (float*)d_out;
  // output layout: logits-logsoftmax (V), context (H), hidden (2H), attn (S)
  float* out_logits = out;
  float* out_ctx    = out + Vn;
  float* out_h0     = out + Vn + Hn;
  float* out_h1     = out + Vn + 2 * Hn;
  float* out_attn   = out + Vn + 3 * Hn;

  // workspace layout (floats)
  float* ws       = (float*)d_ws;
  float* rnn_in   = ws;                    // 2048
  float* gi0      = rnn_in   + 2 * Hn;     // 3072
  float* gh0      = gi0      + 3 * Hn;     // 3072
  float* h0       = gh0      + 3 * Hn;     // 1024
  float* gi1      = h0       + Hn;         // 3072
  float* gh1      = gi1      + 3 * Hn;     // 3072
  float* h1       = gh1      + 3 * Hn;     // 1024  } contiguous => concat
  float* ctx      = h1       + Hn;         // 1024  } [h1, context]
  float* vvec     = ctx      + Hn;         // 1024
  float* energies = vvec     + Hn;         // 4096
  float* attn     = energies + Sn;         // 4096
  float* apart    = attn     + Sn;         // 16*1024
  float* cpart    = apart    + 16 * Hn;    // 32*1024
  float* logits   = cpart    + 32 * Hn;    // V (padded region follows)
  float* scal     = logits   + ((Vn + 15) / 16) * 16;  // 2 scalars

  // 1. rnn_in = [emb[word], last_context]
  embed_concat<<<(2 * Hn + 255) / 256, 256, 0, stream>>>(word, emb, last_ctx, rnn_in);

  // 2. GRU layer 0
  launch_gemv(W_ih0, rnn_in,       b_ih0, gi0, 3 * Hn, 2 * Hn, stream);
  launch_gemv(W_hh0, last_hid,     b_hh0, gh0, 3 * Hn, Hn,     stream);
  gru_gate<<<(Hn + 255) / 256, 256, 0, stream>>>(gi0, gh0, last_hid, h0, out_h0);

  // 3. GRU layer 1
  launch_gemv(W_ih1, h0,           b_ih1, gi1, 3 * Hn, Hn, stream);
  launch_gemv(W_hh1, last_hid + Hn,b_hh1, gh1, 3 * Hn, Hn, stream);
  gru_gate<<<(Hn + 255) / 256, 256, 0, stream>>>(gi1, gh1, last_hid + Hn, h1, out_h1);

  // 4. v[j] = sum_i h1[i] * W_attn[i, j]   (column GEMV, split-K)
  col_gemv_partial<<<16, Hn, 0, stream>>>(W_attn, h1, apart, Hn, Hn);
  reduce_parts<<<(Hn + 255) / 256, 256, 0, stream>>>(apart, vvec, nullptr, 16, Hn);

  // 5. energies[s] = dot(enc[s,:], v)      (row GEMV on S x H)
  launch_gemv(enc, vvec, nullptr, energies, Sn, Hn, stream);

  // 6. attn = softmax(energies)
  softmax_kernel<<<1, 1024, 0, stream>>>(energies, attn, out_attn, Sn);

  // 7. context[i] = sum_s attn[s] * enc[s, i]   (column GEMV, split-K)
  col_gemv_partial<<<32, Hn, 0, stream>>>(enc, attn, cpart, Sn, Hn);
  reduce_parts<<<(Hn + 255) / 256, 256, 0, stream>>>(cpart, ctx, out_ctx, 32, Hn);

  // 8. logits = [h1, context] @ W_out.T + b_out   (dominant: 412 MB stream)
  launch_gemv(W_out, h1 /* == concat base */, b_out, logits, Vn, 2 * Hn, stream);

  // 9. output = log_softmax(logits)
  logsumexp_kernel<<<1, 1024, 0, stream>>>(logits, scal, Vn);
  log_softmax_write<<<(Vn + 255) / 256, 256, 0, stream>>>(logits, scal, out_logits, Vn);
}